// Edge_Connect_28278064677127
// MI455X (gfx1250) — compile-verified
//
#include <hip/hip_runtime.h>
#include <hip/hip_bf16.h>

// Edge connectivity kernel for MI455X (gfx1250, wave32).
// Pure bandwidth-bound gather/normalize: ~280 MB streamed HBM traffic,
// positions (2.4 MB) pinned in L2 by giving all streamed accesses TH_NT.
// One edge per lane, grid-stride x4 with global_prefetch_b8 lookahead.

#define EC_THREADS 256
#define EC_EPT 4  // edges per thread (grid-stride iterations)

__global__ __launch_bounds__(EC_THREADS)
void edge_connect_kernel(const float* __restrict__ pos,      // [N,3]
                         const int*   __restrict__ row_idx,  // [E]
                         const int*   __restrict__ col_idx,  // [E]
                         const float* __restrict__ shift,    // [E,3]
                         float* __restrict__ out_row,        // [E]
                         float* __restrict__ out_col,        // [E]
                         float* __restrict__ out_dist,       // [E]
                         float* __restrict__ out_vec,        // [E,3]
                         int E)
{
    const int stride = (int)(gridDim.x * blockDim.x);
    for (int i = (int)(blockIdx.x * blockDim.x + threadIdx.x); i < E; i += stride) {
        // Prefetch next grid-stride iteration's streamed data into GL2
        // (emits global_prefetch_b8; speculative, dropped if OOB anyway,
        //  but guard to keep addresses valid).
        const int nxt = i + stride;
        if (nxt < E) {
            __builtin_prefetch(row_idx + nxt, 0, 0);
            __builtin_prefetch(col_idx + nxt, 0, 0);
            __builtin_prefetch(shift + 3ll * nxt, 0, 0);
        }

        // Streamed, read-once data: non-temporal so it never evicts the
        // L2-resident positions array.
        const int r = __builtin_nontemporal_load(row_idx + i);
        const int c = __builtin_nontemporal_load(col_idx + i);
        const float sx = __builtin_nontemporal_load(shift + 3ll * i + 0);
        const float sy = __builtin_nontemporal_load(shift + 3ll * i + 1);
        const float sz = __builtin_nontemporal_load(shift + 3ll * i + 2);

        // Random gathers: default (RT) policy -> positions stays hot in L2.
        const float* pr = pos + 3ll * r;
        const float* pc = pos + 3ll * c;
        float vx = pr[0] - pc[0] - sx;
        float vy = pr[1] - pc[1] - sy;
        float vz = pr[2] - pc[2] - sz;

        // Reference semantics: self-loops -> dist 0, raw (unnormalized) vector.
        float dist = 0.0f;
        if (r != c) {
            const float d = sqrtf(vx * vx + vy * vy + vz * vz);
            dist = d;
            vx /= d;  // exact IEEE divides to match jnp's elementwise divide
            vy /= d;
            vz /= d;
        }

        // Write-once outputs: non-temporal streaming stores.
        __builtin_nontemporal_store((float)r, out_row + i);
        __builtin_nontemporal_store((float)c, out_col + i);
        __builtin_nontemporal_store(dist, out_dist + i);
        __builtin_nontemporal_store(vx, out_vec + 3ll * i + 0);
        __builtin_nontemporal_store(vy, out_vec + 3ll * i + 1);
        __builtin_nontemporal_store(vz, out_vec + 3ll * i + 2);
    }
}

extern "C" void kernel_launch(void* const* d_in, const int* in_sizes, int n_in,
                              void* d_out, int out_size, void* d_ws, size_t ws_size,
                              hipStream_t stream) {
    // Inputs (setup_inputs order): positions f32 [N,3], batch int [N] (unused),
    // edge_indices int [2,E], shift f32 [E,3].
    const float* positions    = (const float*)d_in[0];
    const int*   edge_indices = (const int*)d_in[2];
    const float* shift        = (const float*)d_in[3];

    const int E = in_sizes[2] / 2;

    // Output tuple concatenated flat as float32:
    //   [0, 2E)   edge_indices (row block, then col block)
    //   [2E, 3E)  edge_distances
    //   [3E, 6E)  edge_vectors [E,3]
    float* out      = (float*)d_out;
    float* out_row  = out;
    float* out_col  = out + (size_t)E;
    float* out_dist = out + (size_t)2 * E;
    float* out_vec  = out + (size_t)3 * E;

    const int per_block = EC_THREADS * EC_EPT;
    const int blocks = (E + per_block - 1) / per_block;  // 6250 blocks @ E=6.4M

    edge_connect_kernel<<<dim3(blocks), dim3(EC_THREADS), 0, stream>>>(
        positions, edge_indices, edge_indices + E, shift,
        out_row, out_col, out_dist, out_vec, E);
}